// DeferredMLP_24223615549693
// MI455X (gfx1250) — compile-verified
//
#include <hip/hip_runtime.h>
#include <math.h>

// ---------------- problem constants ----------------
#define NPTS      262144
#define K_FEAT    8
#define NUM_KER   4
#define GRID      8
#define NETW      16
#define DEG_ENC   4
#define C_IN0     35          // 8 + 3 + 24
#define CINP      36          // padded to multiple of 4 for WMMA K-blocks
#define EPSV      1e-7f
#define NBASE     9           // base cell coord in [-1..7] -> 9 values
#define NBINS     (NUM_KER * NBASE * NBASE * NBASE)   // 2916

// ---------------- workspace layout (bytes) ----------------
#define OFF_X0    ((size_t)0)                         // N*36 f32
#define OFF_W8    ((size_t)37748736)                  // N*8  f32
#define OFF_BIN   ((size_t)46137344)                  // N    i32
#define OFF_ENT   ((size_t)47185920)                  // N    i32
#define OFF_CNT   ((size_t)48234496)                  // NBINS i32
#define OFF_OFS   ((size_t)48250880)                  // NBINS i32
#define OFF_CUR   ((size_t)48267264)                  // NBINS i32

typedef __attribute__((ext_vector_type(2))) float v2f;
typedef __attribute__((ext_vector_type(8))) float v8f;

__device__ __forceinline__ v8f wmma4(v2f a, v2f b, v8f c) {
  // V_WMMA_F32_16X16X4_F32 : D = A(16x4) * B(4x16) + C(16x16)
  return __builtin_amdgcn_wmma_f32_16x16x4_f32(
      false, a, false, b, (short)0, c, false, false);
}

// ---------------- kernel 0: zero bin counts ----------------
__global__ void zero_counts(int* __restrict__ cnt) {
  int i = blockIdx.x * blockDim.x + threadIdx.x;
  if (i < NBINS) cnt[i] = 0;
}

// ---------------- kernel 1: per-point prep ----------------
__global__ void prep_kernel(const float* __restrict__ features,
                            const float* __restrict__ viewdirs,
                            const float* __restrict__ origins,
                            const float* __restrict__ exposures,
                            const int*   __restrict__ pidxs,
                            float* __restrict__ X0,
                            float* __restrict__ w8,
                            int*   __restrict__ binId,
                            int*   __restrict__ cnt) {
  int n = blockIdx.x * blockDim.x + threadIdx.x;
  if (n >= NPTS) return;

  float x[CINP];
  float le = logf(exposures[n]);
#pragma unroll
  for (int i = 0; i < K_FEAT; ++i) {
    float f = features[n * K_FEAT + i];
    f = fminf(fmaxf(f, EPSV), 1.0f - EPSV);
    x[i] = logf(f) - logf(1.0f - f) + le;
  }
  float v0 = viewdirs[n * 3 + 0];
  float v1 = viewdirs[n * 3 + 1];
  float v2 = viewdirs[n * 3 + 2];
  x[8] = v0; x[9] = v1; x[10] = v2;
#pragma unroll
  for (int s = 0; s < DEG_ENC; ++s) {
    float sc = (float)(1 << s);
    float xb0 = v0 * sc, xb1 = v1 * sc, xb2 = v2 * sc;
    x[11 + s * 3 + 0] = sinf(xb0);
    x[11 + s * 3 + 1] = sinf(xb1);
    x[11 + s * 3 + 2] = sinf(xb2);
    x[23 + s * 3 + 0] = cosf(xb0);
    x[23 + s * 3 + 1] = cosf(xb1);
    x[23 + s * 3 + 2] = cosf(xb2);
  }
  x[35] = 0.0f;
#pragma unroll
  for (int i = 0; i < CINP; ++i) X0[(size_t)n * CINP + i] = x[i];

  // pos / base cell / trilerp weights
  float l0 = (0.5f * (origins[n * 3 + 0] + 1.0f)) * (float)GRID - 0.5f;
  float l1 = (0.5f * (origins[n * 3 + 1] + 1.0f)) * (float)GRID - 0.5f;
  float l2 = (0.5f * (origins[n * 3 + 2] + 1.0f)) * (float)GRID - 0.5f;
  float fb0 = floorf(l0), fb1 = floorf(l1), fb2 = floorf(l2);
  int b0 = (int)fb0, b1 = (int)fb1, b2 = (int)fb2;
  float c0 = l0 - fb0, c1 = l1 - fb1, c2 = l2 - fb2;
  float f0 = 1.0f - c0, f1 = 1.0f - c1, f2 = 1.0f - c2;
#pragma unroll
  for (int c = 0; c < 8; ++c) {
    int dx = (c >> 2) & 1, dy = (c >> 1) & 1, dz = c & 1;
    float w = (dx ? c0 : f0) * (dy ? c1 : f1) * (dz ? c2 : f2);
    w8[(size_t)n * 8 + c] = w;
  }
  int pi = pidxs[n];
  int bin = ((pi * NBASE + (b2 + 1)) * NBASE + (b1 + 1)) * NBASE + (b0 + 1);
  binId[n] = bin;
  atomicAdd(&cnt[bin], 1);
}

// ---------------- kernel 2: exclusive scan of bin counts ----------------
__global__ void scan_kernel(const int* __restrict__ cnt,
                            int* __restrict__ ofs,
                            int* __restrict__ cur) {
  __shared__ int ssum[256];
  const int CH = 12;  // 256*12 = 3072 >= NBINS
  int t = threadIdx.x;
  int s = 0;
#pragma unroll
  for (int i = 0; i < CH; ++i) {
    int idx = t * CH + i;
    if (idx < NBINS) s += cnt[idx];
  }
  ssum[t] = s;
  __syncthreads();
  for (int off = 1; off < 256; off <<= 1) {
    int v = (t >= off) ? ssum[t - off] : 0;
    __syncthreads();
    ssum[t] += v;
    __syncthreads();
  }
  int run = ssum[t] - s;  // exclusive prefix of this chunk
  for (int i = 0; i < CH; ++i) {
    int idx = t * CH + i;
    if (idx < NBINS) {
      ofs[idx] = run;
      cur[idx] = run;
      run += cnt[idx];
    }
  }
}

// ---------------- kernel 3: scatter points into bins ----------------
__global__ void scatter_kernel(const int* __restrict__ binId,
                               int* __restrict__ cur,
                               int* __restrict__ entries) {
  int n = blockIdx.x * blockDim.x + threadIdx.x;
  if (n >= NPTS) return;
  int slot = atomicAdd(&cur[binId[n]], 1);
  entries[slot] = n;
}

// ---------------- kernel 4: fused 3-layer WMMA MLP, one block per bin ----
// 3 waves/SIMD floor => ~336 VGPR cap: enough headroom for the software-
// pipelined tile loop so nothing spills to scratch (demand is ~270 regs).
__global__ void __launch_bounds__(128, 3)
mlp_kernel(const float* __restrict__ X0, const float* __restrict__ w8,
           const int* __restrict__ entries, const int* __restrict__ ofs,
           const int* __restrict__ cnt,
           const float* __restrict__ W1, const float* __restrict__ b1,
           const float* __restrict__ W2, const float* __restrict__ b2,
           const float* __restrict__ W3, const float* __restrict__ b3,
           float* __restrict__ out) {
  __shared__ float sW1[8][CINP][16];   // 18.4 KB (k=35 row zero-padded)
  __shared__ float sW2[8][16][16];     // 8 KB
  __shared__ float sW3[8][16][16];     // 8 KB (cols 3..15 zero)
  __shared__ float sB1[8][16], sB2[8][16], sB3[8][16];
  __shared__ float sYt[4][16][16];     // per-wave transpose buffer

  int bin = blockIdx.x;
  int count = cnt[bin];
  if (count == 0) return;
  int base = ofs[bin];

  // decode bin -> kernel index + base cell coords
  int pi = bin / (NBASE * NBASE * NBASE);
  int r  = bin - pi * (NBASE * NBASE * NBASE);
  int bz = r / (NBASE * NBASE) - 1; r %= (NBASE * NBASE);
  int by = r / NBASE - 1;
  int bx = r % NBASE - 1;

  int cellOff[8];
#pragma unroll
  for (int c = 0; c < 8; ++c) {
    int dx = (c >> 2) & 1, dy = (c >> 1) & 1, dz = c & 1;
    int xi = bx + dx; xi = xi < 0 ? 0 : (xi > 7 ? 7 : xi);
    int yi = by + dy; yi = yi < 0 ? 0 : (yi > 7 ? 7 : yi);
    int zi = bz + dz; zi = zi < 0 ? 0 : (zi > 7 ? 7 : zi);
    cellOff[c] = pi * 512 + (zi * 8 + yi) * 8 + xi;
  }

  int tid = threadIdx.x;
  // cooperative load of the 8 corner weight sets into LDS
  for (int e = tid; e < 8 * CINP * 16; e += 128) {
    int c = e / (CINP * 16);
    int rem = e - c * (CINP * 16);
    int k = rem >> 4, d = rem & 15;
    float v = 0.0f;
    if (k < C_IN0) v = W1[((size_t)cellOff[c] * C_IN0 + k) * 16 + d];
    sW1[c][k][d] = v;
  }
  for (int e = tid; e < 8 * 16 * 16; e += 128) {
    int c = e >> 8, k = (e >> 4) & 15, d = e & 15;
    sW2[c][k][d] = W2[((size_t)cellOff[c] * 16 + k) * 16 + d];
    sW3[c][k][d] = (d < 3) ? W3[((size_t)cellOff[c] * 16 + k) * 3 + d] : 0.0f;
  }
  for (int e = tid; e < 8 * 16; e += 128) {
    int c = e >> 4, d = e & 15;
    sB1[c][d] = b1[(size_t)cellOff[c] * 16 + d];
    sB2[c][d] = b2[(size_t)cellOff[c] * 16 + d];
    sB3[c][d] = (d < 3) ? b3[(size_t)cellOff[c] * 3 + d] : 0.0f;
  }
  __syncthreads();

  int wave = tid >> 5;
  int lane = tid & 31;
  int row  = lane & 15;   // A-fragment M / B-fragment N (col)
  int hi   = lane >> 4;   // K-half select
  int ntiles = (count + 15) >> 4;

  for (int t = wave; t < ntiles; t += 4) {
    // Kill cross-iteration LICM of the LDS weight fragments: without this the
    // compiler hoists ~280 VGPRs of loop-invariant B fragments out of the tile
    // loop, overflows the register file, and spills to scratch. Re-reading LDS
    // per tile is far cheaper than scratch traffic.
    asm volatile("" ::: "memory");

    int ridx = t * 16 + row;
    bool valid = ridx < count;
    int pt = entries[base + (valid ? ridx : 0)];
    float wv[8];
    {
      const float* wp = w8 + (size_t)pt * 8;
#pragma unroll
      for (int c = 0; c < 8; ++c) wv[c] = valid ? wp[c] : 0.0f;
    }

    // ---------------- layer 1: [16x36] x [36x16] (+8 corners) -----------
    v8f acc = {0.f, 0.f, 0.f, 0.f, 0.f, 0.f, 0.f, 0.f};
    // bias as a K=8 GEMM over the corner dimension
#pragma unroll
    for (int kb = 0; kb < 2; ++kb) {
      v2f a, b;
      a.x = hi ? wv[kb * 4 + 2] : wv[kb * 4 + 0];
      a.y = hi ? wv[kb * 4 + 3] : wv[kb * 4 + 1];
      int kr = kb * 4 + 2 * hi;
      b.x = sB1[kr][row];
      b.y = sB1[kr + 1][row];
      acc = wmma4(a, b, acc);
    }
    const float* xrow = X0 + (size_t)pt * CINP;
#pragma unroll
    for (int kb = 0; kb < CINP / 4; ++kb) {
      int k0 = kb * 4 + 2 * hi;
      float2 xv = *reinterpret_cast<const float2*>(xrow + k0);
#pragma unroll
      for (int c = 0; c < 8; ++c) {
        v2f a = {xv.x * wv[c], xv.y * wv[c]};
        v2f b = {sW1[c][k0][row], sW1[c][k0 + 1][row]};
        acc = wmma4(a, b, acc);
      }
    }
    // relu + transpose (C lane l holds y[v+8*hi][row])
#pragma unroll
    for (int v = 0; v < 8; ++v) {
      float y = acc[v];
      sYt[wave][v + 8 * hi][row] = y > 0.0f ? y : 0.0f;
    }

    // ---------------- layer 2: [16x16] x [16x16] (+8 corners) -----------
    v8f acc2 = {0.f, 0.f, 0.f, 0.f, 0.f, 0.f, 0.f, 0.f};
#pragma unroll
    for (int kb = 0; kb < 2; ++kb) {
      v2f a, b;
      a.x = hi ? wv[kb * 4 + 2] : wv[kb * 4 + 0];
      a.y = hi ? wv[kb * 4 + 3] : wv[kb * 4 + 1];
      int kr = kb * 4 + 2 * hi;
      b.x = sB2[kr][row];
      b.y = sB2[kr + 1][row];
      acc2 = wmma4(a, b, acc2);
    }
#pragma unroll
    for (int kb = 0; kb < 4; ++kb) {
      int k0 = kb * 4 + 2 * hi;
      float xa = sYt[wave][row][k0];
      float xb_ = sYt[wave][row][k0 + 1];
#pragma unroll
      for (int c = 0; c < 8; ++c) {
        v2f a = {xa * wv[c], xb_ * wv[c]};
        v2f b = {sW2[c][k0][row], sW2[c][k0 + 1][row]};
        acc2 = wmma4(a, b, acc2);
      }
    }
#pragma unroll
    for (int v = 0; v < 8; ++v) {
      float y = acc2[v];
      sYt[wave][v + 8 * hi][row] = y > 0.0f ? y : 0.0f;
    }

    // ---------------- layer 3: [16x16] x [16x16(3 used)] ----------------
    v8f acc3 = {0.f, 0.f, 0.f, 0.f, 0.f, 0.f, 0.f, 0.f};
#pragma unroll
    for (int kb = 0; kb < 2; ++kb) {
      v2f a, b;
      a.x = hi ? wv[kb * 4 + 2] : wv[kb * 4 + 0];
      a.y = hi ? wv[kb * 4 + 3] : wv[kb * 4 + 1];
      int kr = kb * 4 + 2 * hi;
      b.x = sB3[kr][row];
      b.y = sB3[kr + 1][row];
      acc3 = wmma4(a, b, acc3);
    }
#pragma unroll
    for (int kb = 0; kb < 4; ++kb) {
      int k0 = kb * 4 + 2 * hi;
      float xa = sYt[wave][row][k0];
      float xb_ = sYt[wave][row][k0 + 1];
#pragma unroll
      for (int c = 0; c < 8; ++c) {
        v2f a = {xa * wv[c], xb_ * wv[c]};
        v2f b = {sW3[c][k0][row], sW3[c][k0 + 1][row]};
        acc3 = wmma4(a, b, acc3);
      }
    }
    // sigmoid + transpose + store (3 output channels)
#pragma unroll
    for (int v = 0; v < 8; ++v) {
      float y = acc3[v];
      sYt[wave][v + 8 * hi][row] = 1.0f / (1.0f + expf(-y));
    }
    if (valid && hi == 0) {
      float* o = out + (size_t)pt * 3;
      o[0] = sYt[wave][row][0];
      o[1] = sYt[wave][row][1];
      o[2] = sYt[wave][row][2];
    }
  }
}

// ---------------- launcher ----------------
extern "C" void kernel_launch(void* const* d_in, const int* in_sizes, int n_in,
                              void* d_out, int out_size, void* d_ws, size_t ws_size,
                              hipStream_t stream) {
  const float* features  = (const float*)d_in[0];
  const float* viewdirs  = (const float*)d_in[1];
  const float* origins   = (const float*)d_in[2];
  const float* exposures = (const float*)d_in[3];
  const int*   pidxs     = (const int*)d_in[4];
  const float* W1 = (const float*)d_in[5];
  const float* b1 = (const float*)d_in[6];
  const float* W2 = (const float*)d_in[7];
  const float* b2 = (const float*)d_in[8];
  const float* W3 = (const float*)d_in[9];
  const float* b3 = (const float*)d_in[10];
  float* out = (float*)d_out;

  char* ws = (char*)d_ws;
  float* X0      = (float*)(ws + OFF_X0);
  float* w8      = (float*)(ws + OFF_W8);
  int*   binId   = (int*)(ws + OFF_BIN);
  int*   entries = (int*)(ws + OFF_ENT);
  int*   cntb    = (int*)(ws + OFF_CNT);
  int*   ofs     = (int*)(ws + OFF_OFS);
  int*   cur     = (int*)(ws + OFF_CUR);

  zero_counts<<<(NBINS + 255) / 256, 256, 0, stream>>>(cntb);
  prep_kernel<<<NPTS / 256, 256, 0, stream>>>(features, viewdirs, origins,
                                              exposures, pidxs, X0, w8, binId, cntb);
  scan_kernel<<<1, 256, 0, stream>>>(cntb, ofs, cur);
  scatter_kernel<<<NPTS / 256, 256, 0, stream>>>(binId, cur, entries);
  mlp_kernel<<<NBINS, 128, 0, stream>>>(X0, w8, entries, ofs, cntb,
                                        W1, b1, W2, b2, W3, b3, out);
}